// TensorProductConv_26663156973855
// MI455X (gfx1250) — compile-verified
//
#include <hip/hip_runtime.h>

typedef __attribute__((ext_vector_type(2))) float v2f;
typedef __attribute__((ext_vector_type(8))) float v8f;

// ---------------------------------------------------------------------------
// Kernel 1a/1b: build CSR rowptr from the sorted `rows` array into d_ws.
// rowptr[r] = first edge whose row == r (or start of next segment for empty
// rows); rowptr[V] = E.  Deterministic, no atomics.
// ---------------------------------------------------------------------------
__global__ void rowptr_init_kernel(int* __restrict__ rowptr, int V, int E) {
    int r = blockIdx.x * blockDim.x + threadIdx.x;
    if (r <= V) rowptr[r] = E;   // covers trailing empty rows + sentinel
}

__global__ void rowptr_fill_kernel(const int* __restrict__ rows,
                                   int* __restrict__ rowptr, int E) {
    int e = blockIdx.x * blockDim.x + threadIdx.x;
    if (e >= E) return;
    int r     = rows[e];
    int rprev = (e == 0) ? -1 : rows[e - 1];
    // Segment boundary: fill this row's start and any empty rows before it.
    for (int q = rprev + 1; q <= r; ++q) rowptr[q] = e;
}

// ---------------------------------------------------------------------------
// Main kernel: one wave32 per (node, 16-channel tile).
//
// Per edge e of the node:
//   M[i,k]      = sum_j Y[e,j] * cg[i,j,k]                (3x9, lane-resident)
//   A_p[c,i]    = X[cols[e], c, i] * W[e, c, p]           (16x4, K=3 padded 0)
//   B_p[i,k]    = M[i,k] * (path(k) == p)                 (4x16, cols 9..15 = 0)
//   acc        += A_p x B_p   for p = 0,1,2               (V_WMMA_F32_16X16X4_F32)
// After all edges: acc[v][k] (k<9) is out[node, tile*16 + v + 8*(lane>=16), k].
// EXEC is all-1s everywhere a WMMA executes (node check is wave-uniform).
// ---------------------------------------------------------------------------
__global__ __launch_bounds__(256) void tpconv_wmma_kernel(
    const float* __restrict__ X,    // [V, 64*3]
    const float* __restrict__ Y,    // [E, 3]
    const float* __restrict__ W,    // [E, 64*3]
    const float* __restrict__ cg,   // [3,3,9]
    const int*   __restrict__ cols, // [E]
    const int*   __restrict__ rowptr, // [V+1]
    float*       __restrict__ out,  // [V, 64*9]
    int V)
{
    const int lane = threadIdx.x & 31;
    const int wave = threadIdx.x >> 5;
    const int task = blockIdx.x * 8 + wave;     // (node, tile) pair
    const int node = task >> 2;
    const int tile = task & 3;
    if (node >= V) return;                      // wave-uniform exit

    const int  k       = lane & 15;            // N index (output component)
    const bool lowHalf = lane < 16;
    const int  i_lo    = lowHalf ? 0 : 2;      // B row held in b[0] by this lane

    // Preload the CG rows this lane needs (b[0] holds row i_lo, b[1] holds
    // row 1 for lanes<16 and the zero pad row K=3 for lanes>=16).
    float cgl0 = 0.f, cgl1 = 0.f, cgl2 = 0.f;
    float cgh0 = 0.f, cgh1 = 0.f, cgh2 = 0.f;
    if (k < 9) {
        cgl0 = cg[(i_lo * 3 + 0) * 9 + k];
        cgl1 = cg[(i_lo * 3 + 1) * 9 + k];
        cgl2 = cg[(i_lo * 3 + 2) * 9 + k];
        if (lowHalf) {
            cgh0 = cg[(1 * 3 + 0) * 9 + k];
            cgh1 = cg[(1 * 3 + 1) * 9 + k];
            cgh2 = cg[(1 * 3 + 2) * 9 + k];
        }
    }
    // path(k): k==0 -> 0 ; k in 1..3 -> 1 ; k in 4..8 -> 2 ; k>=9 -> none
    float m0 = (k == 0) ? 1.f : 0.f;
    float m1 = (k >= 1 && k <= 3) ? 1.f : 0.f;
    float m2 = (k >= 4 && k <= 8) ? 1.f : 0.f;

    const int tbase = tile * 16;
    const int c     = tbase + k;               // channel this lane loads (<64)

    const int e0 = rowptr[node];
    const int e1 = rowptr[node + 1];

    v8f acc = {};                              // 16x16 f32 C/D tile

    for (int e = e0; e < e1; ++e) {
        // Edge geometry (wave-uniform -> scalarizes to s_loads).
        const float y0 = Y[e * 3 + 0];
        const float y1 = Y[e * 3 + 1];
        const float y2 = Y[e * 3 + 2];
        const int   col = cols[e];

        // Software prefetch of the next edge's gathered X row.
        if (e + 1 < e1) {
            const int coln = cols[e + 1];
            __builtin_prefetch(X + (long)coln * 192 + c * 3, 0, 1);
        }

        // M rows for this lane's B slots.
        const float Mlo = y0 * cgl0 + y1 * cgl1 + y2 * cgl2;
        const float Mhi = y0 * cgh0 + y1 * cgh1 + y2 * cgh2;

        // Gather sender features and per-edge weights for channel c.
        const float* xr = X + (long)col * 192 + c * 3;
        const float  xa = lowHalf ? xr[0] : xr[2];   // A K-slot in a[0]
        const float  xb = lowHalf ? xr[1] : 0.f;     // A K-slot in a[1] (K=3 pad)
        const float* wr = W + (long)e * 192 + c * 3;
        const float  w0 = wr[0], w1 = wr[1], w2 = wr[2];

        v2f a, b;
        // path 0 (scalar output, k==0)
        a[0] = xa * w0; a[1] = xb * w0;
        b[0] = Mlo * m0; b[1] = Mhi * m0;
        acc = __builtin_amdgcn_wmma_f32_16x16x4_f32(
            false, a, false, b, (short)0, acc, false, false);
        // path 1 (vector output, k==1..3)
        a[0] = xa * w1; a[1] = xb * w1;
        b[0] = Mlo * m1; b[1] = Mhi * m1;
        acc = __builtin_amdgcn_wmma_f32_16x16x4_f32(
            false, a, false, b, (short)0, acc, false, false);
        // path 2 (l=2 output, k==4..8)
        a[0] = xa * w2; a[1] = xb * w2;
        b[0] = Mlo * m2; b[1] = Mhi * m2;
        acc = __builtin_amdgcn_wmma_f32_16x16x4_f32(
            false, a, false, b, (short)0, acc, false, false);
    }

    // Store: lane holds D rows M = v + 8*(lane>=16), col k.  Only k<9 is real.
    if (k < 9) {
        float* orow = out + (long)node * 576 + (tbase + (lowHalf ? 0 : 8)) * 9 + k;
#pragma unroll
        for (int v = 0; v < 8; ++v) orow[v * 9] = acc[v];
    }
}

// ---------------------------------------------------------------------------
extern "C" void kernel_launch(void* const* d_in, const int* in_sizes, int n_in,
                              void* d_out, int out_size, void* d_ws, size_t ws_size,
                              hipStream_t stream) {
    const float* X    = (const float*)d_in[0];  // [V, C*3]
    const float* Y    = (const float*)d_in[1];  // [E, 3]
    const float* W    = (const float*)d_in[2];  // [E, C*3]
    const float* cg   = (const float*)d_in[3];  // [3,3,9]
    const int*   rows = (const int*)d_in[4];    // [E] sorted
    const int*   cols = (const int*)d_in[5];    // [E]

    const int V = in_sizes[0] / 192;            // C*L1_DIM = 192
    const int E = in_sizes[1] / 3;

    int* rowptr = (int*)d_ws;                   // (V+1) ints of scratch

    rowptr_init_kernel<<<(V + 256) / 256, 256, 0, stream>>>(rowptr, V, E);
    rowptr_fill_kernel<<<(E + 255) / 256, 256, 0, stream>>>(rows, rowptr, E);

    const int tasks  = V * 4;                   // (node, channel-tile) waves
    const int blocks = (tasks + 7) / 8;         // 8 waves per 256-thread block
    tpconv_wmma_kernel<<<blocks, 256, 0, stream>>>(
        X, Y, W, cg, cols, rowptr, (float*)d_out, V);
}